// BahdanauAttention_85916525789819
// MI455X (gfx1250) — compile-verified
//
#include <hip/hip_runtime.h>
#include <stdint.h>

typedef unsigned int u32;
typedef unsigned long long u64;
typedef u32 v4u __attribute__((ext_vector_type(4)));
typedef int v8i __attribute__((ext_vector_type(8)));
typedef int v4i __attribute__((ext_vector_type(4)));

// Problem dims (from the reference)
#define BB 32
#define TT 2048
#define DD 1024

// TDM tile: 8192 x 2 fp32 elements = 64 KiB per workgroup chunk
#define TILE0 8192u
#define TILE1 2u
#define CHUNK (TILE0 * TILE1) // 16384 floats

// ---------------------------------------------------------------------------
// Bulk copy context_vector = values via the Tensor Data Mover (gfx1250).
// One wave per workgroup builds a D# and DMAs global->LDS->global.
// D# layout per CDNA5 ISA.
// ---------------------------------------------------------------------------
__launch_bounds__(32)
__global__ void BahdanauAttention_tdm_copy(const float* __restrict__ src,
                                           float* __restrict__ dst) {
    __shared__ float lds_buf[CHUNK]; // 64 KiB staging tile

    const u64 base = (u64)blockIdx.x * (u64)CHUNK;
    const u64 ga_src = (u64)(uintptr_t)(src + base);
    const u64 ga_dst = (u64)(uintptr_t)(dst + base);
    // Flat LDS aperture: low 32 bits of the generic pointer are the LDS byte
    // offset (CDNA5 ISA aperture mapping).
    const u32 lds_addr = (u32)(uintptr_t)(void*)&lds_buf[0];

    // --- D# group 0 (128b): count=1 | lds_addr | global_addr[56:0] | type=2
    v4u g0_load = { 1u,
                    lds_addr,
                    (u32)ga_src,
                    (u32)((ga_src >> 32) & 0x01FFFFFFu) | 0x80000000u };
    v4u g0_store = { 1u,
                     lds_addr,
                     (u32)ga_dst,
                     (u32)((ga_dst >> 32) & 0x01FFFFFFu) | 0x80000000u };

    // --- D# group 1 (256b):
    // [17:16] data_size=2 (4B); mask/pad/iterate = 0
    // [79:48]  tensor_dim0 = TILE0   (== tile -> no OOB clamping)
    // [111:80] tensor_dim1 = TILE1
    // [127:112] tile_dim0 = TILE0
    // [143:128] tile_dim1 = TILE1 ; tile_dim2 = 0
    // [207:160] tensor_dim0_stride = TILE0 (contiguous rows)
    v8i g1 = { (int)(2u << 16),
               (int)((TILE0 & 0xFFFFu) << 16),
               (int)((TILE0 >> 16) | ((TILE1 & 0xFFFFu) << 16)),
               (int)((TILE1 >> 16) | (TILE0 << 16)),
               (int)(TILE1 & 0xFFFFu),
               (int)TILE0,
               0, 0 };
    v4i gz4 = { 0, 0, 0, 0 };
    v8i gz8 = { 0, 0, 0, 0, 0, 0, 0, 0 };

    // DMA global -> LDS (TENSORcnt++), wait, DMA LDS -> global, wait.
    // 6-arg builtin form (clang-23 lane): (g0, g1, g2, g3, gx, cpol).
    __builtin_amdgcn_tensor_load_to_lds(g0_load, g1, gz4, gz4, gz8, 0);
    __builtin_amdgcn_s_wait_tensorcnt(0);
    __builtin_amdgcn_tensor_store_from_lds(g0_store, g1, gz4, gz4, gz8, 0);
    __builtin_amdgcn_s_wait_tensorcnt(0);
}

// ---------------------------------------------------------------------------
// attention_weights[b,t,0] = 1.0 (softmax over a size-1 axis is identically 1)
// coverage_vectors[b,t,0]  = t   (running sum of ones, starting at 0)
// ---------------------------------------------------------------------------
__global__ void BahdanauAttention_fill_aw_cov(float* __restrict__ att,
                                              float* __restrict__ cov,
                                              int n) {
    int i = blockIdx.x * blockDim.x + threadIdx.x;
    if (i < n) {
        att[i] = 1.0f;
        cov[i] = (float)(i & (TT - 1)); // t = i % T, T is a power of two
    }
}

extern "C" void kernel_launch(void* const* d_in, const int* in_sizes, int n_in,
                              void* d_out, int out_size, void* d_ws, size_t ws_size,
                              hipStream_t stream) {
    (void)in_sizes; (void)n_in; (void)d_ws; (void)ws_size; (void)out_size;

    // Inputs (setup_inputs order): 0=query, 1=values, 2..9 = weights/biases.
    // All weights are dead code: softmax over a singleton axis forces a_t == 1.
    const float* values = (const float*)d_in[1];

    const size_t ctx_elems = (size_t)BB * TT * DD; // 67,108,864
    const size_t bt = (size_t)BB * TT;             // 65,536

    float* ctx = (float*)d_out;        // context_vector = values
    float* att = ctx + ctx_elems;      // attention_weights = 1
    float* cov = att + bt;             // coverage_vectors = t

    // 256 MB copy via TDM: 4096 workgroups x 64 KiB tiles, 1 wave each.
    const unsigned nblocks = (unsigned)(ctx_elems / CHUNK); // 4096, exact
    BahdanauAttention_tdm_copy<<<nblocks, 32, 0, stream>>>(values, ctx);

    // Tiny tail fills (512 KB total).
    BahdanauAttention_fill_aw_cov<<<(unsigned)((bt + 255) / 256), 256, 0, stream>>>(
        att, cov, (int)bt);
}